// GANLoss_52321291600268
// MI455X (gfx1250) — compile-verified
//
#include <hip/hip_runtime.h>
#include <hip/hip_bf16.h>

typedef __attribute__((ext_vector_type(2))) float v2f;
typedef __attribute__((ext_vector_type(8))) float v8f;

// Exact-f32 wave32 all-reduce using V_WMMA_F32_16X16X4_F32.
// A (16x4 f32, 2 VGPRs/lane): lane value in component 0, zero in component 1.
// B (4x16 f32, 2 VGPRs/lane): all ones (layout-independent).
// D[m][n] = sum_k A[m][k]  (row sums, replicated across columns).
// Per lane: sum of the 8 D VGPRs = sum over rows 0..7 (lanes 0-15) or rows
// 8..15 (lanes 16-31); one xor-16 shuffle + add yields the full 32-lane sum
// in every lane. Must be executed with all 32 lanes of the wave active.
__device__ __forceinline__ float wave32_allreduce_wmma(float v) {
    v2f a;
    a[0] = v;
    a[1] = 0.0f;
    v2f b;
    b[0] = 1.0f;
    b[1] = 1.0f;
    v8f c = {};
    v8f d = __builtin_amdgcn_wmma_f32_16x16x4_f32(
        /*neg_a=*/false, a, /*neg_b=*/false, b,
        /*c_mod=*/(short)0, c, /*reuse_a=*/false, /*reuse_b=*/false);
    float s = d[0] + d[1] + d[2] + d[3] + d[4] + d[5] + d[6] + d[7];
    s += __shfl_xor(s, 16, 32);
    return s;
}

// Stage 1: one gather per thread, block reduce to a single partial per block.
__global__ void __launch_bounds__(256)
gather_partial_kernel(const float* __restrict__ prob,
                      const long long* __restrict__ targets,
                      const float* __restrict__ reward,
                      float* __restrict__ partial,
                      int N, long long C) {
    __shared__ float sdata[256];
    const int t = threadIdx.x;
    const int n = blockIdx.x * 256 + t;

    float p = 0.0f;
    if (n < N) {
        const long long tgt = targets[n];
        p = prob[(long long)n * C + tgt] * reward[n];
    }
    sdata[t] = p;
    __syncthreads();

    if (t < 32) {  // wave 0 only, full 32-lane EXEC -> WMMA legal
        float acc = 0.0f;
#pragma unroll
        for (int j = 0; j < 8; ++j) {
            acc += sdata[t + 32 * j];
        }
        const float s = wave32_allreduce_wmma(acc);
        if (t == 0) {
            partial[blockIdx.x] = s;
        }
    }
}

// Stage 2: single wave reduces the per-block partials, writes -mean.
__global__ void __launch_bounds__(32)
final_reduce_kernel(const float* __restrict__ partial,
                    float* __restrict__ out,
                    int n_partials, float inv_n) {
    const int t = threadIdx.x;  // exactly 32 threads -> full EXEC
    float acc = 0.0f;
    for (int i = t; i < n_partials; i += 32) {
        acc += partial[i];
    }
    const float s = wave32_allreduce_wmma(acc);
    if (t == 0) {
        out[0] = -s * inv_n;
    }
}

extern "C" void kernel_launch(void* const* d_in, const int* in_sizes, int n_in,
                              void* d_out, int out_size, void* d_ws, size_t ws_size,
                              hipStream_t stream) {
    const float* prob        = (const float*)d_in[0];       // (N, C) f32
    const long long* targets = (const long long*)d_in[1];   // (N,)  int64
    const float* reward      = (const float*)d_in[2];       // (N,)  f32
    float* out               = (float*)d_out;               // scalar f32

    const int N = in_sizes[1];                      // 8192
    const long long C = (long long)(in_sizes[0] / in_sizes[1]);  // 32000

    const int blocks = (N + 255) / 256;             // 32
    float* partial = (float*)d_ws;                  // blocks * 4 bytes scratch

    gather_partial_kernel<<<blocks, 256, 0, stream>>>(prob, targets, reward,
                                                      partial, N, C);
    final_reduce_kernel<<<1, 32, 0, stream>>>(partial, out, blocks,
                                              1.0f / (float)N);
}